// Propagate_33208687133411
// MI455X (gfx1250) — compile-verified
//
#include <hip/hip_runtime.h>

// K-hop graph propagation (CSR SpMM, fixed degree 16) for MI455X / gfx1250.
//
// Per row r: out[r,:] = sum_{k<16} w[r,k] * x[nbr[r,k], :]  (D_FEAT = 64, f32)
// One wave32 per row. Accumulation runs on V_WMMA_F32_16X16X4_F32 using a
// broadcast-A trick: all 16 M-rows of A carry the same 4 edge weights, B holds
// 4 gathered neighbor rows x 16-feature slice. 4 K-steps x 4 slices = 16 WMMAs
// per row; D's rows are identical, so acc[t][0] is the output slice.
//
// f32 16x16x4 operand layout (ISA 7.12.2, wave32):
//   A (16x4): lanes 0-15 = M; VGPR0 holds K0 (lo half) / K2 (hi half), VGPR1 K1/K3
//   B (4x16): lanes carry N; VGPR0 holds K0-row (lo) / K2-row (hi), VGPR1 K1/K3
//   C/D (16x16): VGPR v, lanes 0-15 -> M=v,N=lane; lanes 16-31 -> M=v+8,N=lane-16

typedef float v2f __attribute__((ext_vector_type(2)));
typedef float v8f __attribute__((ext_vector_type(8)));

#define GP_N_NODES 100000
#define GP_DEG 16
#define GP_D_FEAT 64

__global__ __launch_bounds__(256) void spmm_hop_wmma(
    const float* __restrict__ xin,     // [N, 64] f32, L2-resident (25.6 MB)
    const float* __restrict__ vals,    // [N*16]  f32 edge weights
    const int*   __restrict__ indices, // [N*16]  neighbor ids
    float*       __restrict__ xout,    // [N, 64] f32
    int nrows)
{
    const int  lane = threadIdx.x & 31;
    const bool hi   = lane >= 16;     // upper half-wave carries K=2,3 / M=8..15
    const int  fl   = lane & 15;      // feature lane within a 16-wide slice
    const int  wavesPerBlock = blockDim.x >> 5;
    const int  wave   = blockIdx.x * wavesPerBlock + (threadIdx.x >> 5);
    const int  stride = gridDim.x * wavesPerBlock;

    for (int row = wave; row < nrows; row += stride) {
        // Row id is wave-uniform: force SGPR so weight/index reads become
        // scalar loads (KMcnt path), keeping VMEM free for feature gathers.
        const int r = __builtin_amdgcn_readfirstlane(row);
        const float* __restrict__ wp = vals    + (size_t)r * GP_DEG;
        const int*   __restrict__ ip = indices + (size_t)r * GP_DEG;

        v2f A[4];        // [k-step] broadcast weights
        v2f B[4][4];     // [k-step][feature-slice] gathered neighbor data

        #pragma unroll
        for (int s = 0; s < 4; ++s) {
            const float w0 = wp[4*s+0], w1 = wp[4*s+1];
            const float w2 = wp[4*s+2], w3 = wp[4*s+3];
            const int   n0 = ip[4*s+0], n1 = ip[4*s+1];
            const int   n2 = ip[4*s+2], n3 = ip[4*s+3];

            // A[m,k] = w[k] for every m (row-broadcast weights)
            A[s].x = hi ? w2 : w0;   // VGPR0: K0 / K2
            A[s].y = hi ? w3 : w1;   // VGPR1: K1 / K3

            // Two gather base addresses per step; each load is 32 lanes x 4B
            // hitting two contiguous 64B row-slices (fully coalesced in L2).
            const float* bx = xin + (size_t)(hi ? n2 : n0) * GP_D_FEAT + fl;
            const float* by = xin + (size_t)(hi ? n3 : n1) * GP_D_FEAT + fl;
            #pragma unroll
            for (int t = 0; t < 4; ++t) {
                B[s][t].x = bx[16 * t];  // immediate-offset global_load_b32
                B[s][t].y = by[16 * t];
            }
        }

        // 16 chained f32 WMMAs; the 4 slice-accumulators are independent for ILP.
        v8f acc[4] = {};
        #pragma unroll
        for (int s = 0; s < 4; ++s) {
            #pragma unroll
            for (int t = 0; t < 4; ++t) {
                acc[t] = __builtin_amdgcn_wmma_f32_16x16x4_f32(
                    /*neg_a=*/false, A[s], /*neg_b=*/false, B[s][t],
                    /*c_mod=*/(short)0, acc[t],
                    /*reuse_a=*/false, /*reuse_b=*/false);
            }
        }

        // All D rows are identical; VGPR0 of acc[t] holds slice t at N=lane&15.
        // Pack into two fully coalesced 128B stores per row.
        float* __restrict__ op = xout + (size_t)row * GP_D_FEAT;
        op[lane]      = hi ? acc[1][0] : acc[0][0];  // features  0..31
        op[32 + lane] = hi ? acc[3][0] : acc[2][0];  // features 32..63
    }
}

extern "C" void kernel_launch(void* const* d_in, const int* in_sizes, int n_in,
                              void* d_out, int out_size, void* d_ws, size_t ws_size,
                              hipStream_t stream) {
    (void)in_sizes; (void)n_in; (void)out_size; (void)ws_size;

    const float* x       = (const float*)d_in[0];  // [N,64] f32
    const float* values  = (const float*)d_in[1];  // [E]    f32
    /* d_in[2] = indptr: unused, degree is fixed at 16 */
    const int*   indices = (const int*)  d_in[3];  // [E]    int
    float* out = (float*)d_out;                    // [N,64] f32
    float* ws  = (float*)d_ws;                     // >= 25.6 MB scratch

    const int nrows       = GP_N_NODES;
    const int threads     = 256;                   // 8 waves per block
    const int rowsPerWave = 4;
    const int waves  = (nrows + rowsPerWave - 1) / rowsPerWave;
    const int blocks = (waves + (threads / 32) - 1) / (threads / 32);

    // 3 hops, stream-ordered (graph-capture safe): x -> out -> ws -> out
    spmm_hop_wmma<<<blocks, threads, 0, stream>>>(x,   values, indices, out, nrows);
    spmm_hop_wmma<<<blocks, threads, 0, stream>>>(out, values, indices, ws,  nrows);
    spmm_hop_wmma<<<blocks, threads, 0, stream>>>(ws,  values, indices, out, nrows);
}